// QuantumLayer_MultiQunode_56315611185631
// MI455X (gfx1250) — compile-verified
//
#include <hip/hip_runtime.h>
#include <cstddef>

typedef float v2f __attribute__((ext_vector_type(2)));
typedef float v8f __attribute__((ext_vector_type(8)));

#define NSTATE 20736   // 12^4
#define ENT    20736   // 144*144
#define NW     4       // waves per workgroup in main kernel

// ---------------- workspace layout (float offsets) ----------------
#define GATE1_SZ 576ull                      // 3 planes * (16 rows * 12 cols)
#define ENC_OFF  0ull                        // 4096 per-sample displacement gates
#define SQ_OFF   (ENC_OFF + 4096ull*GATE1_SZ)
#define DP_OFF   (SQ_OFF  + 8ull*GATE1_SZ)
#define DIAG_OFF (DP_OFF  + 8ull*GATE1_SZ)   // 24 vectors * 12 complex
#define BS_OFF   (DIAG_OFF + 576ull)         // 24 gates * 3 planes * 144*144
#define BS_GSZ   (3ull*ENT)
#define BSW_OFF  (BS_OFF + 24ull*BS_GSZ)     // expm work: 4 complex bufs per gate
#define BSW_SZ   (8ull*ENT)

// ---------------- helpers ----------------
__device__ __forceinline__ int st1(int m) {   // complex stride of mode m
  return m == 0 ? 1728 : (m == 1 ? 144 : (m == 2 ? 12 : 1));
}
__device__ __forceinline__ int st2(int m) {   // stride of combined (m,m+1) index
  return m == 0 ? 144 : (m == 1 ? 12 : 1);
}

__device__ __forceinline__ v8f wmma_f32(v2f a, v2f bb, v8f c) {
  // D = A(16x4,f32) * B(4x16,f32) + C(16x16,f32); neg_a/neg_b must be 0 for f32
  return __builtin_amdgcn_wmma_f32_16x16x4_f32(false, a, false, bb, (short)0, c,
                                               false, false);
}

// ============ prep kernel 1: 12x12 expm gates (disp encode / squeeze / disp) ============
__global__ void prep_gates12(const float* __restrict__ x,
                             const float* __restrict__ rr_, const float* __restrict__ phr_,
                             const float* __restrict__ aa_, const float* __restrict__ pha_,
                             float* __restrict__ ws) {
  __shared__ float2 A[144], P[144], T[144], U[144];
  const int t = threadIdx.x;
  const int id = blockIdx.x;
  const int row = t / 12, col = t % 12;

  float rr, ph; int kind; float* out;
  if (id < 4096) {                       // amplitude/phase encoding: Displacement(x[b,m], x[b,m+4])
    int b = id >> 2, m = id & 3;
    rr = x[b * 8 + m]; ph = x[b * 8 + 4 + m]; kind = 0;
    out = ws + ENC_OFF + (size_t)id * GATE1_SZ;
  } else if (id < 4104) {                // squeeze gates
    int q = id - 4096;
    rr = rr_[q]; ph = phr_[q]; kind = 1;
    out = ws + SQ_OFF + (size_t)q * GATE1_SZ;
  } else {                               // layer displacement gates
    int q = id - 4104;
    rr = aa_[q]; ph = pha_[q]; kind = 0;
    out = ws + DP_OFF + (size_t)q * GATE1_SZ;
  }

  if (t < 144) {
    float ar = rr * cosf(ph), ai = rr * sinf(ph);
    float gre = 0.f, gim = 0.f;
    if (kind == 0) {                     // alpha*adag - conj(alpha)*a
      if (col == row - 1) { float s = sqrtf((float)row); gre =  ar * s; gim =  ai * s; }
      else if (col == row + 1) { float s = sqrtf((float)col); gre = -ar * s; gim =  ai * s; }
    } else {                             // 0.5*(conj(z)*a^2 - z*(a^2)^dag)
      if (col == row + 2) { float s = 0.5f * sqrtf((float)((row + 1) * (row + 2))); gre =  ar * s; gim = -ai * s; }
      else if (col == row - 2) { float s = 0.5f * sqrtf((float)(row * (row - 1)));  gre = -ar * s; gim = -ai * s; }
    }
    const float sc = 1.0f / 64.0f;       // scaling 2^-6
    float2 g; g.x = gre * sc; g.y = gim * sc;
    A[t] = g; P[t] = g;
    float2 tt = g; if (row == col) tt.x += 1.0f;
    T[t] = tt;
  }
  __syncthreads();
  for (int j = 2; j <= 12; ++j) {        // Taylor order 12
    if (t < 144) {
      float sr = 0.f, si = 0.f;
      for (int k = 0; k < 12; ++k) {
        float2 av = A[row * 12 + k], pv = P[k * 12 + col];
        sr += av.x * pv.x - av.y * pv.y;
        si += av.x * pv.y + av.y * pv.x;
      }
      float inv = 1.0f / (float)j;
      float2 u; u.x = sr * inv; u.y = si * inv; U[t] = u;
    }
    __syncthreads();
    if (t < 144) { P[t] = U[t]; T[t].x += U[t].x; T[t].y += U[t].y; }
    __syncthreads();
  }
  for (int q = 0; q < 6; ++q) {          // squaring 2^6
    if (t < 144) {
      float sr = 0.f, si = 0.f;
      for (int k = 0; k < 12; ++k) {
        float2 av = T[row * 12 + k], bv = T[k * 12 + col];
        sr += av.x * bv.x - av.y * bv.y;
        si += av.x * bv.y + av.y * bv.x;
      }
      float2 u; u.x = sr; u.y = si; U[t] = u;
    }
    __syncthreads();
    if (t < 144) T[t] = U[t];
    __syncthreads();
  }
  // planes: Ur | Ui | -Ui, padded to 16 rows of 12 cols (rows 12..15 zero)
  for (int idx = t; idx < 192; idx += blockDim.x) {
    float vr = 0.f, vi = 0.f;
    if (idx < 144) { vr = T[idx].x; vi = T[idx].y; }
    out[idx] = vr; out[192 + idx] = vi; out[384 + idx] = -vi;
  }
}

// ============ prep kernel 2: rotation / Kerr diagonals ============
__global__ void prep_diags(const float* __restrict__ vp1, const float* __restrict__ vp2,
                           const float* __restrict__ kap, float* __restrict__ ws) {
  int t = threadIdx.x;
  if (t >= 288) return;
  int vec = t / 12, nn = t % 12;
  float phase;
  if (vec < 8)       phase = vp1[vec] * (float)nn;
  else if (vec < 16) phase = vp2[vec - 8] * (float)nn;
  else               phase = kap[vec - 16] * (float)(nn * nn);
  float* d = ws + DIAG_OFF + (size_t)vec * 24;
  d[nn * 2] = cosf(phase);
  d[nn * 2 + 1] = sinf(phase);
}

// ============ prep kernel 3: 144x144 beamsplitter expm (global scratch) ============
__global__ void prep_bs(const float* __restrict__ th1, const float* __restrict__ ph1,
                        const float* __restrict__ th2, const float* __restrict__ ph2,
                        float* __restrict__ ws) {
  const int g = blockIdx.x;                 // 0..23 : (l, interferometer, j)
  const int l = g / 12, intf = (g % 12) / 6, j = g % 6;
  const float th = (intf == 0 ? th1 : th2)[l * 6 + j];
  const float ph = (intf == 0 ? ph1 : ph2)[l * 6 + j];
  const float cr = th * cosf(ph), ci = th * sinf(ph);
  float* wk = ws + BSW_OFF + (size_t)g * BSW_SZ;
  float* Ab = wk;               // scaled generator
  float* Pb = wk + 2 * ENT;     // Taylor power term
  float* Tb = wk + 4 * ENT;     // accumulated exp
  float* Ub = wk + 6 * ENT;     // temp
  const int tid = threadIdx.x, nt = blockDim.x;
  const float sc = 1.0f / 1024.0f;          // scaling 2^-10

  for (int e = tid; e < ENT; e += nt) {
    int rI = e / 144, cI = e % 144;
    int i = rI / 12, jm = rI % 12, kq = cI / 12, lm = cI % 12;
    float hre = 0.f, him = 0.f;
    if (kq == i + 1 && lm == jm - 1) {      // th*e^{i ph} * a1 a2^dag
      float s = sqrtf((float)((i + 1) * jm)); hre += cr * s; him += ci * s;
    }
    if (kq == i - 1 && lm == jm + 1) {      // -th*e^{-i ph} * a1^dag a2
      float s = sqrtf((float)(i * (jm + 1))); hre -= cr * s; him += ci * s;
    }
    hre *= sc; him *= sc;
    Ab[2 * e] = hre; Ab[2 * e + 1] = him;
    Pb[2 * e] = hre; Pb[2 * e + 1] = him;
    Tb[2 * e] = hre + (rI == cI ? 1.0f : 0.0f); Tb[2 * e + 1] = him;
  }
  __syncthreads();

  float* Pp = Pb; float* Up = Ub;
  for (int jj = 2; jj <= 12; ++jj) {        // Taylor
    for (int e = tid; e < ENT; e += nt) {
      int rI = e / 144, cI = e % 144;
      float sr = 0.f, si = 0.f;
      for (int k = 0; k < 144; ++k) {
        float ar = Ab[2 * (rI * 144 + k)], ai = Ab[2 * (rI * 144 + k) + 1];
        float pr = Pp[2 * (k * 144 + cI)], pi = Pp[2 * (k * 144 + cI) + 1];
        sr += ar * pr - ai * pi;
        si += ar * pi + ai * pr;
      }
      float inv = 1.0f / (float)jj;
      Up[2 * e] = sr * inv; Up[2 * e + 1] = si * inv;
    }
    __syncthreads();
    for (int e = tid; e < ENT; e += nt) { Tb[2 * e] += Up[2 * e]; Tb[2 * e + 1] += Up[2 * e + 1]; }
    __syncthreads();
    float* tmp = Pp; Pp = Up; Up = tmp;
  }
  float* Sq = Ab;                           // generator no longer needed
  for (int q = 0; q < 10; ++q) {            // squaring 2^10
    for (int e = tid; e < ENT; e += nt) {
      int rI = e / 144, cI = e % 144;
      float sr = 0.f, si = 0.f;
      for (int k = 0; k < 144; ++k) {
        float ar = Tb[2 * (rI * 144 + k)], ai = Tb[2 * (rI * 144 + k) + 1];
        float br = Tb[2 * (k * 144 + cI)], bi = Tb[2 * (k * 144 + cI) + 1];
        sr += ar * br - ai * bi;
        si += ar * bi + ai * br;
      }
      Sq[2 * e] = sr; Sq[2 * e + 1] = si;
    }
    __syncthreads();
    for (int e = tid; e < ENT; e += nt) { Tb[2 * e] = Sq[2 * e]; Tb[2 * e + 1] = Sq[2 * e + 1]; }
    __syncthreads();
  }
  float* outp = ws + BS_OFF + (size_t)g * BS_GSZ;   // planes Ur | Ui | -Ui (144x144)
  for (int e = tid; e < ENT; e += nt) {
    float vr = Tb[2 * e], vi = Tb[2 * e + 1];
    outp[e] = vr; outp[ENT + e] = vi; outp[2 * ENT + e] = -vi;
  }
}

// ============ main circuit kernel: state resident in LDS, WMMA f32 contractions ============

// 1-mode gate (12x12 complex) along stride s; in-place, dual fiber tiles per pass.
// A-tiles (9 x v2f) are hoisted into VGPRs for all 27 tiles owned by this wave.
__device__ __noinline__ void apply_gate12(float* sre, float* sim,
                                          const float* __restrict__ g, int s,
                                          int wave, int lane) {
  const int cl = lane & 15, hi = lane >> 4;
  v2f aur[3], aui[3], aun[3];
#pragma unroll
  for (int k4 = 0; k4 < 3; ++k4) {
    int ao = cl * 12 + k4 * 4 + hi * 2;
    aur[k4] = *(const v2f*)(g + ao);
    aui[k4] = *(const v2f*)(g + 192 + ao);
    aun[k4] = *(const v2f*)(g + 384 + ao);
  }
  const int base = wave * 27;               // this wave owns fiber tiles [base, base+27)
  for (int i = 0; i < 14; ++i) {            // 13 pairs + 1 single
    int ft0 = base + 2 * i;
    int ft1 = (i < 13) ? (ft0 + 1) : ft0;
    int f0 = ft0 * 16 + cl, f1 = ft1 * 16 + cl;
    int fib0 = (f0 / s) * (12 * s) + (f0 % s);
    int fib1 = (f1 / s) * (12 * s) + (f1 % s);
    v8f aR0 = {0.f,0.f,0.f,0.f,0.f,0.f,0.f,0.f};
    v8f aI0 = {0.f,0.f,0.f,0.f,0.f,0.f,0.f,0.f};
    v8f aR1 = {0.f,0.f,0.f,0.f,0.f,0.f,0.f,0.f};
    v8f aI1 = {0.f,0.f,0.f,0.f,0.f,0.f,0.f,0.f};
#pragma unroll
    for (int k4 = 0; k4 < 3; ++k4) {
      int c0 = k4 * 4 + hi * 2;
      int a0 = fib0 + c0 * s, a1 = fib1 + c0 * s;
      v2f br0 = {sre[a0], sre[a0 + s]};
      v2f bi0 = {sim[a0], sim[a0 + s]};
      v2f br1 = {sre[a1], sre[a1 + s]};
      v2f bi1 = {sim[a1], sim[a1 + s]};
      aR0 = wmma_f32(aur[k4], br0, aR0);   // Ur*psi_r
      aR0 = wmma_f32(aun[k4], bi0, aR0);   // -Ui*psi_i
      aI0 = wmma_f32(aur[k4], bi0, aI0);   // Ur*psi_i
      aI0 = wmma_f32(aui[k4], br0, aI0);   // Ui*psi_r
      aR1 = wmma_f32(aur[k4], br1, aR1);
      aR1 = wmma_f32(aun[k4], bi1, aR1);
      aI1 = wmma_f32(aur[k4], bi1, aI1);
      aI1 = wmma_f32(aui[k4], br1, aI1);
    }
#pragma unroll
    for (int v = 0; v < 8; ++v) {
      int rowo = hi * 8 + v;
      if (rowo < 12) {
        sre[fib0 + rowo * s] = aR0[v]; sim[fib0 + rowo * s] = aI0[v];
        sre[fib1 + rowo * s] = aR1[v]; sim[fib1 + rowo * s] = aI1[v];
      }
    }
  }
}

// 2-mode gate (144x144 complex) along stride s; dual fiber tiles share each A-load.
// Pass 0: tiles (wave, wave+4); pass 1: tile 8 by wave 0 only.
__device__ __noinline__ void apply_gate144(float* sre, float* sim,
                                           const float* __restrict__ g, int s,
                                           float* tw, int wave, int lane) {
  const float* Ur = g;
  const float* Ui = g + ENT;
  const float* Un = g + 2 * ENT;
  const int cl = lane & 15, hi = lane >> 4;
  float* t0re = tw;
  float* t0im = tw + 2304;
  float* t1re = tw + 4608;
  float* t1im = tw + 6912;
  for (int pass = 0; pass < 2; ++pass) {
    if (pass == 1 && wave != 0) break;      // wave-uniform branch
    int ft0 = (pass == 0) ? wave : 8;
    int ft1 = (pass == 0) ? wave + 4 : 8;
    int f0 = ft0 * 16 + cl, f1 = ft1 * 16 + cl;
    int fib0 = (f0 / s) * (144 * s) + (f0 % s);
    int fib1 = (f1 / s) * (144 * s) + (f1 % s);
#pragma unroll 1
    for (int mt = 0; mt < 9; ++mt) {
      v8f aR0 = {0.f,0.f,0.f,0.f,0.f,0.f,0.f,0.f};
      v8f aI0 = {0.f,0.f,0.f,0.f,0.f,0.f,0.f,0.f};
      v8f aR1 = {0.f,0.f,0.f,0.f,0.f,0.f,0.f,0.f};
      v8f aI1 = {0.f,0.f,0.f,0.f,0.f,0.f,0.f,0.f};
#pragma unroll 2
      for (int k4 = 0; k4 < 36; ++k4) {
        int ao = (mt * 16 + cl) * 144 + k4 * 4 + hi * 2;
        v2f ur = *(const v2f*)(Ur + ao);
        v2f ui = *(const v2f*)(Ui + ao);
        v2f un = *(const v2f*)(Un + ao);
        int c0 = k4 * 4 + hi * 2;
        int a0 = fib0 + c0 * s, a1 = fib1 + c0 * s;
        v2f br0 = {sre[a0], sre[a0 + s]};
        v2f bi0 = {sim[a0], sim[a0 + s]};
        v2f br1 = {sre[a1], sre[a1 + s]};
        v2f bi1 = {sim[a1], sim[a1 + s]};
        aR0 = wmma_f32(ur, br0, aR0);
        aR0 = wmma_f32(un, bi0, aR0);
        aI0 = wmma_f32(ur, bi0, aI0);
        aI0 = wmma_f32(ui, br0, aI0);
        aR1 = wmma_f32(ur, br1, aR1);
        aR1 = wmma_f32(un, bi1, aR1);
        aI1 = wmma_f32(ur, bi1, aI1);
        aI1 = wmma_f32(ui, br1, aI1);
      }
#pragma unroll
      for (int v = 0; v < 8; ++v) {
        int rowo = mt * 16 + hi * 8 + v;
        t0re[rowo * 16 + cl] = aR0[v]; t0im[rowo * 16 + cl] = aI0[v];
        t1re[rowo * 16 + cl] = aR1[v]; t1im[rowo * 16 + cl] = aI1[v];
      }
    }
    // write both 144 x 16 output blocks back into the state (this wave's fibers only)
    for (int e = lane; e < 2304; e += 32) {
      int rr = e >> 4, cc = e & 15;
      int fa = ft0 * 16 + cc;
      int fba = (fa / s) * (144 * s) + (fa % s);
      sre[fba + rr * s] = t0re[e]; sim[fba + rr * s] = t0im[e];
      int fb = ft1 * 16 + cc;
      int fbb = (fb / s) * (144 * s) + (fb % s);
      sre[fbb + rr * s] = t1re[e]; sim[fbb + rr * s] = t1im[e];
    }
  }
}

__device__ __noinline__ void apply_diag(float* sre, float* sim,
                                        const float* __restrict__ d, int s,
                                        int tid, float* stage) {
  if (tid < 24) stage[tid] = d[tid];
  __syncthreads();
  for (int e = tid; e < NSTATE; e += 128) {
    int dig = (e / s) % 12;
    float pr = stage[2 * dig], pi = stage[2 * dig + 1];
    float ar = sre[e], ai = sim[e];
    sre[e] = ar * pr - ai * pi;
    sim[e] = ar * pi + ai * pr;
  }
}

__global__ __launch_bounds__(128) void qcircuit(const float* __restrict__ ws,
                                                float* __restrict__ out) {
  extern __shared__ float smem[];
  float* sre  = smem;                 // 20736 floats
  float* sim  = smem + NSTATE;        // 20736 floats
  float* temp = smem + 2 * NSTATE;    // 4 waves * 9216 floats (dual tile staging)
  const int tid = threadIdx.x;
  const int wave = tid >> 5, lane = tid & 31;
  const int b = blockIdx.x;
  float* tw = temp + wave * 9216;

  for (int e = tid; e < 2 * NSTATE; e += 128) smem[e] = 0.0f;
  __syncthreads();
  if (tid == 0) sre[0] = 1.0f;        // vacuum |0000>
  __syncthreads();

  // --- amplitude/phase encoding: per-sample displacement on each mode ---
  const float* enc = ws + ENC_OFF + (size_t)(b * 4) * GATE1_SZ;
#pragma unroll
  for (int m = 0; m < 4; ++m) {
    apply_gate12(sre, sim, enc + (size_t)m * GATE1_SZ, st1(m), wave, lane);
    __syncthreads();
  }

  const int mmArr[6] = {0, 2, 1, 0, 2, 1};  // Clements mesh mode order

  for (int l = 0; l < 2; ++l) {
    // interferometer 1
#pragma unroll
    for (int j = 0; j < 6; ++j) {
      int mo = mmArr[j];
      apply_gate144(sre, sim, ws + BS_OFF + (size_t)((l * 2 + 0) * 6 + j) * BS_GSZ,
                    st2(mo), tw, wave, lane);
      __syncthreads();
    }
#pragma unroll
    for (int m = 0; m < 4; ++m) {            // varphi_1 rotations
      apply_diag(sre, sim, ws + DIAG_OFF + (size_t)(l * 4 + m) * 24, st1(m), tid, temp);
      __syncthreads();
    }
#pragma unroll
    for (int m = 0; m < 4; ++m) {            // squeezers
      apply_gate12(sre, sim, ws + SQ_OFF + (size_t)(l * 4 + m) * GATE1_SZ, st1(m), wave, lane);
      __syncthreads();
    }
    // interferometer 2
#pragma unroll
    for (int j = 0; j < 6; ++j) {
      int mo = mmArr[j];
      apply_gate144(sre, sim, ws + BS_OFF + (size_t)((l * 2 + 1) * 6 + j) * BS_GSZ,
                    st2(mo), tw, wave, lane);
      __syncthreads();
    }
#pragma unroll
    for (int m = 0; m < 4; ++m) {            // varphi_2 rotations
      apply_diag(sre, sim, ws + DIAG_OFF + (size_t)(8 + l * 4 + m) * 24, st1(m), tid, temp);
      __syncthreads();
    }
#pragma unroll
    for (int m = 0; m < 4; ++m) {            // displacements
      apply_gate12(sre, sim, ws + DP_OFF + (size_t)(l * 4 + m) * GATE1_SZ, st1(m), wave, lane);
      __syncthreads();
    }
#pragma unroll
    for (int m = 0; m < 4; ++m) {            // Kerr
      apply_diag(sre, sim, ws + DIAG_OFF + (size_t)(16 + l * 4 + m) * 24, st1(m), tid, temp);
      __syncthreads();
    }
  }

  // --- X-quadrature expectation values: X = a + a^dag (tridiagonal) ---
  float* red = temp;
#pragma unroll
  for (int m = 0; m < 4; ++m) {
    int s = st1(m);
    float local = 0.f;
    for (int e = tid; e < NSTATE; e += 128) {
      int dig = (e / s) % 12;
      float xr = 0.f, xi = 0.f;
      if (dig < 11) { float q = sqrtf((float)(dig + 1)); xr += q * sre[e + s]; xi += q * sim[e + s]; }
      if (dig > 0)  { float q = sqrtf((float)dig);       xr += q * sre[e - s]; xi += q * sim[e - s]; }
      local += sre[e] * xr + sim[e] * xi;   // Re(conj(psi) * X psi)
    }
    red[tid] = local;
    __syncthreads();
    if (tid == 0) {
      float tot = 0.f;
      for (int q = 0; q < 128; ++q) tot += red[q];
      out[b * 4 + m] = tot;
    }
    __syncthreads();
  }
}

// ---------------- host side ----------------
extern "C" void kernel_launch(void* const* d_in, const int* in_sizes, int n_in,
                              void* d_out, int out_size, void* d_ws, size_t ws_size,
                              hipStream_t stream) {
  (void)in_sizes; (void)n_in; (void)out_size; (void)ws_size;
  const float* x   = (const float*)d_in[0];
  const float* th1 = (const float*)d_in[1];
  const float* ph1 = (const float*)d_in[2];
  const float* vp1 = (const float*)d_in[3];
  const float* r   = (const float*)d_in[4];
  const float* phr = (const float*)d_in[5];
  const float* th2 = (const float*)d_in[6];
  const float* ph2 = (const float*)d_in[7];
  const float* vp2 = (const float*)d_in[8];
  const float* a   = (const float*)d_in[9];
  const float* pha = (const float*)d_in[10];
  const float* kap = (const float*)d_in[11];
  float* ws  = (float*)d_ws;
  float* out = (float*)d_out;

  prep_gates12<<<4112, 160, 0, stream>>>(x, r, phr, a, pha, ws);
  prep_diags<<<1, 288, 0, stream>>>(vp1, vp2, kap, ws);
  prep_bs<<<24, 256, 0, stream>>>(th1, ph1, th2, ph2, ws);

  const size_t shbytes = (size_t)(2 * NSTATE + 4 * 9216) * sizeof(float); // 313,344 B < 320 KB WGP LDS
  (void)hipFuncSetAttribute(reinterpret_cast<const void*>(qcircuit),
                            hipFuncAttributeMaxDynamicSharedMemorySize, (int)shbytes);
  qcircuit<<<1024, 128, shbytes, stream>>>(ws, out);
}